// Q_Online_58342835749707
// MI455X (gfx1250) — compile-verified
//
#include <hip/hip_runtime.h>
#include <hip/hip_bf16.h>

typedef float v2f __attribute__((ext_vector_type(2)));
typedef float v8f __attribute__((ext_vector_type(8)));

#define NIN_C    12
#define NHID_C   512
#define NU_C     40
#define NINEQ_C  148
#define BATCH_C  4096
#define ITERS_C  20
#define SIGMA_C  0.1f
#define BIG_C    1.0e9f
#define EPS_C    1.0e-4f

// Pitch 49 (odd, coprime with 64 banks): conflict-free row walks across lanes
// and column walks (stride 49) in Cholesky. Columns 40..48 are zero padding so
// WMMA fragment loads/stores for the 48-wide tile grid need no guards.
#define QS 49
#define GS 49
#define HS 49

__device__ __forceinline__ void wsync() { __builtin_amdgcn_wave_barrier(); }

__device__ __forceinline__ float wred_sum(float v) {
#pragma unroll
  for (int m = 16; m >= 1; m >>= 1) v += __shfl_xor(v, m, 32);
  return v;
}
__device__ __forceinline__ float wred_min(float v) {
#pragma unroll
  for (int m = 16; m >= 1; m >>= 1) v = fminf(v, __shfl_xor(v, m, 32));
  return v;
}

// ---------------------------------------------------------------------------
// Kernel 1: build Q_hat [40x40], G [148x40], h [148]  (tiny, one workgroup)
// ---------------------------------------------------------------------------
__global__ __launch_bounds__(128) void build_qp_kernel(
    const float* __restrict__ L, const float* __restrict__ LP,
    const float* __restrict__ LR, const float* __restrict__ A,
    const float* __restrict__ Bm, const float* __restrict__ u0,
    const float* __restrict__ s0, float* __restrict__ ws) {
  __shared__ float Qx[12][12], Pm[12][12], Rm[4][4];
  __shared__ float sA[12][12];
  __shared__ float blk[9][12][4];     // A^r * B
  __shared__ float Bh[108][40];       // B_hat
  __shared__ float Mi[108][40];       // Q_diag @ B_hat
  const int tid = threadIdx.x;

  for (int o = tid; o < 144; o += 128) sA[o / 12][o % 12] = A[o];
  for (int o = tid; o < 144; o += 128) {
    int i = o / 12, j = o % 12;
    int m = (i < j) ? i : j;
    float a = 0.f, b = 0.f;
    for (int k = 0; k <= m; ++k) {
      a += L[i * 12 + k] * L[j * 12 + k];
      b += LP[i * 12 + k] * LP[j * 12 + k];
    }
    Qx[i][j] = a + ((i == j) ? EPS_C : 0.f);
    Pm[i][j] = b + ((i == j) ? EPS_C : 0.f);
  }
  for (int o = tid; o < 16; o += 128) {
    int i = o / 4, j = o % 4;
    int m = (i < j) ? i : j;
    float a = 0.f;
    for (int k = 0; k <= m; ++k) a += LR[i * 4 + k] * LR[j * 4 + k];
    Rm[i][j] = a + ((i == j) ? EPS_C : 0.f);
  }
  for (int o = tid; o < 48; o += 128) blk[0][o / 4][o % 4] = Bm[o];
  __syncthreads();

  for (int r = 1; r < 9; ++r) {
    for (int o = tid; o < 48; o += 128) {
      int i = o / 4, c = o % 4;
      float a = 0.f;
      for (int k = 0; k < 12; ++k) a += sA[i][k] * blk[r - 1][k][c];
      blk[r][i][c] = a;
    }
    __syncthreads();
  }
  // B_hat block(r,c) = A^(r-c) B for c<=r else 0   (r=0..8, c=0..9)
  for (int o = tid; o < 108 * 40; o += 128) {
    int row = o / 40, col = o % 40;
    int rb = row / 12, ri = row % 12, cb = col / 4, cj = col % 4;
    Bh[row][col] = (cb <= rb) ? blk[rb - cb][ri][cj] : 0.f;
  }
  __syncthreads();
  // Mi = block_diag(Qx x8, P) @ B_hat
  for (int o = tid; o < 108 * 40; o += 128) {
    int row = o / 40, col = o % 40;
    int rb = row / 12, ri = row % 12;
    float a = 0.f;
    if (rb < 8)
      for (int k = 0; k < 12; ++k) a += Qx[ri][k] * Bh[rb * 12 + k][col];
    else
      for (int k = 0; k < 12; ++k) a += Pm[ri][k] * Bh[rb * 12 + k][col];
    Mi[row][col] = a;
  }
  __syncthreads();

  float* gQ = ws;                       // 1600
  float* gG = ws + NU_C * NU_C;         // 5920
  float* gh = gG + NINEQ_C * NU_C;      // 148
  for (int o = tid; o < NU_C * NU_C; o += 128) {
    int i = o / 40, k = o % 40;
    float a = 0.f;
    for (int j = 0; j < 108; ++j) a += Bh[j][i] * Mi[j][k];
    if ((i >> 2) == (k >> 2)) a += Rm[i & 3][k & 3];
    gQ[o] = a;
  }
  for (int o = tid; o < NINEQ_C * NU_C; o += 128) {
    int j = o / 40, i = o % 40;
    gG[o] = (j < 40) ? ((j == i) ? 1.f : 0.f) : Bh[j - 40][i];
  }
  for (int j = tid; j < NINEQ_C; j += 128) {
    float a = s0[j];
    if (j < 40) a += u0[j];
    else
      for (int i = 0; i < 40; ++i) a += Bh[j - 40][i] * u0[i];
    gh[j] = a;
  }
}

// ---------------------------------------------------------------------------
// Kernel 2: fused MLP, p = leaky(leaky(x W1^T + b1) W2^T + b2)   [4096 x 40]
// ---------------------------------------------------------------------------
__global__ __launch_bounds__(256) void mlp_kernel(
    const float* __restrict__ x, const float* __restrict__ W1,
    const float* __restrict__ b1, const float* __restrict__ W2,
    const float* __restrict__ b2, float* __restrict__ p_out) {
  extern __shared__ float sm[];
  float* hdn = sm;                 // 32 rows x 513 pitch
  float* xs = sm + 32 * 513;       // 32 x 12
  const int tid = threadIdx.x;
  const int b0 = blockIdx.x * 32;

  for (int o = tid; o < 32 * NIN_C; o += 256) xs[o] = x[b0 * NIN_C + o];
  __syncthreads();
  for (int o = tid; o < 32 * NHID_C; o += 256) {
    int r = o >> 9, c = o & 511;
    float a = b1[c];
    const float* w = &W1[c * NIN_C];
    const float* xr = &xs[r * NIN_C];
    for (int k = 0; k < NIN_C; ++k) a += xr[k] * w[k];
    hdn[r * 513 + c] = (a > 0.f) ? a : 0.01f * a;
  }
  __syncthreads();
  for (int o = tid; o < 32 * NU_C; o += 256) {
    int r = o / NU_C, u = o % NU_C;
    float a = b2[u];
    const float* w = &W2[u * NHID_C];
    const float* hr = &hdn[r * 513];
    for (int k = 0; k < NHID_C; ++k) a += hr[k] * w[k];
    p_out[(b0 + r) * NU_C + u] = (a > 0.f) ? a : 0.01f * a;
  }
}

// ---------------------------------------------------------------------------
// Kernel 3: full 20-iteration IPM. One wave32 per batch element, 4 per WG.
// H = Q + G^T diag(d) G built with V_WMMA_F32_16X16X4_F32 (3x3 tiles, K=148),
// C initialized from the (zero-padded) Q tile so the add is fused into the
// first accumulate; all fragment loads/stores unconditional via padding.
// ---------------------------------------------------------------------------
__global__ __launch_bounds__(128) void ipm_kernel(
    const float* __restrict__ qp, const float* __restrict__ pglob,
    float* __restrict__ out) {
  extern __shared__ float sm[];
  float* sQ = sm;                        // 48*49 (zero-padded to 48x48 tiles)
  float* sG = sQ + 48 * QS;              // 148*49 (cols 40..48 zero)
  float* sh = sG + NINEQ_C * GS;         // 148
  float* ebase = sh + NINEQ_C;

  const int tid = threadIdx.x;
  const int wave = tid >> 5;
  const int lane = tid & 31;
  const int lh = lane & 15;
  const int sel = lane >> 4;             // K half (0: K=0,1  1: K=2,3)

  const float* gQ = qp;
  const float* gG = qp + NU_C * NU_C;
  const float* gh = gG + NINEQ_C * NU_C;
  // zero padded regions first, then fill real data
  for (int o = tid; o < 48 * QS; o += 128) sQ[o] = 0.f;
  for (int o = tid; o < NINEQ_C * GS; o += 128) sG[o] = 0.f;
  __syncthreads();
  for (int o = tid; o < NU_C * NU_C; o += 128)
    sQ[(o / NU_C) * QS + (o % NU_C)] = gQ[o];
  for (int o = tid; o < NINEQ_C * NU_C; o += 128)
    sG[(o / NU_C) * GS + (o % NU_C)] = gG[o];
  for (int o = tid; o < NINEQ_C; o += 128) sh[o] = gh[o];
  __syncthreads();

  const int EF = 48 * HS + 7 * NINEQ_C + 6 * NU_C;     // floats per element
  float* E = ebase + wave * EF;
  float* Hm  = E;                 // 48*49 (only 40x40 region consumed)
  float* sv  = Hm + 48 * HS;      // s     [148]
  float* lv  = sv + NINEQ_C;      // lam
  float* dvv = lv + NINEQ_C;      // d
  float* tvv = dvv + NINEQ_C;     // (rc+lam*rp)/s
  float* rpv = tvv + NINEQ_C;     // rp
  float* dsv = rpv + NINEQ_C;     // ds
  float* dlv = dsv + NINEQ_C;     // dlam
  float* uv  = dlv + NINEQ_C;     // u     [40]
  float* pv  = uv + NU_C;
  float* rdv = pv + NU_C;
  float* rhsv = rdv + NU_C;
  float* duv = rhsv + NU_C;
  float* yv  = duv + NU_C;

  const int b = blockIdx.x * 4 + wave;

  for (int j = lane; j < NINEQ_C; j += 32) { sv[j] = 1.f; lv[j] = 1.f; }
  for (int i = lane; i < NU_C; i += 32) { uv[i] = 0.f; pv[i] = pglob[b * NU_C + i]; }
  wsync();

  for (int it = 0; it < ITERS_C; ++it) {
    // rd = u@Q + p + lam@G
    for (int i = lane; i < NU_C; i += 32) {
      float a = pv[i];
      for (int k = 0; k < NU_C; ++k) a += uv[k] * sQ[k * QS + i];
      for (int j = 0; j < NINEQ_C; ++j) a += lv[j] * sG[j * GS + i];
      rdv[i] = a;
    }
    // rp = u@G^T + s - h
    for (int j = lane; j < NINEQ_C; j += 32) {
      float a = sv[j] - sh[j];
      const float* gr = &sG[j * GS];
      for (int i = 0; i < NU_C; ++i) a += uv[i] * gr[i];
      rpv[j] = a;
    }
    // mu = mean(s*lam)
    float part = 0.f;
    for (int j = lane; j < NINEQ_C; j += 32) part += sv[j] * lv[j];
    const float mu = wred_sum(part) * (1.0f / NINEQ_C);
    // d = lam/s ; t = (rc + lam*rp)/s, rc = sigma*mu - s*lam
    for (int j = lane; j < NINEQ_C; j += 32) {
      float sj = sv[j], lj = lv[j];
      float rc = SIGMA_C * mu - sj * lj;
      dvv[j] = lj / sj;
      tvv[j] = (rc + lj * rpv[j]) / sj;
    }
    wsync();
    // rhs = -(rd + t@G)
    for (int i = lane; i < NU_C; i += 32) {
      float a = rdv[i];
      for (int j = 0; j < NINEQ_C; ++j) a += tvv[j] * sG[j * GS + i];
      rhsv[i] = -a;
    }

    // H = Q + G^T diag(d) G via fp32 WMMA 16x16x4 (3x3 tiles, K = 148 = 37*4)
    v8f acc[3][3];
#pragma unroll
    for (int ti = 0; ti < 3; ++ti)
#pragma unroll
      for (int tk = 0; tk < 3; ++tk)
#pragma unroll
        for (int r = 0; r < 8; ++r)
          acc[ti][tk][r] = sQ[(ti * 16 + r + sel * 8) * QS + tk * 16 + lh];

#pragma clang loop unroll(disable)
    for (int j0 = 0; j0 < NINEQ_C; j0 += 4) {
      const int ja = j0 + (sel << 1);      // this lane-half's K rows
      const float d0 = dvv[ja];
      const float d1 = dvv[ja + 1];
      const float* g0 = &sG[ja * GS + lh];
      const float* g1 = g0 + GS;
      v2f raw[3], af[3];
#pragma unroll
      for (int t = 0; t < 3; ++t) {
        const float r0 = g0[t * 16];         // zero in pad columns
        const float r1 = g1[t * 16];
        raw[t][0] = r0; raw[t][1] = r1;          // B fragment: G[j][col]
        af[t][0] = r0 * d0; af[t][1] = r1 * d1;  // A fragment: G[j][col]*d[j]
      }
#pragma unroll
      for (int ti = 0; ti < 3; ++ti)
#pragma unroll
        for (int tk = 0; tk < 3; ++tk)
          acc[ti][tk] = __builtin_amdgcn_wmma_f32_16x16x4_f32(
              false, af[ti], false, raw[tk], (short)0, acc[ti][tk], false, false);
    }
#pragma unroll
    for (int ti = 0; ti < 3; ++ti)
#pragma unroll
      for (int tk = 0; tk < 3; ++tk)
#pragma unroll
        for (int r = 0; r < 8; ++r)
          Hm[(ti * 16 + r + sel * 8) * HS + tk * 16 + lh] = acc[ti][tk][r];
    wsync();

    // Cholesky (in place, lower)
    for (int k = 0; k < NU_C; ++k) {
      const float lkk = sqrtf(Hm[k * HS + k]);
      const float inv = 1.0f / lkk;
      if (lane == 0) Hm[k * HS + k] = lkk;
      for (int i = k + 1 + lane; i < NU_C; i += 32) Hm[i * HS + k] *= inv;
      wsync();
      for (int i = k + 1 + lane; i < NU_C; i += 32) {
        const float lik = Hm[i * HS + k];
        for (int j = k + 1; j <= i; ++j) Hm[i * HS + j] -= lik * Hm[j * HS + k];
      }
      wsync();
    }
    // forward substitution: L y = rhs
    for (int i = 0; i < NU_C; ++i) {
      float p2 = 0.f;
      for (int j = lane; j < i; j += 32) p2 += Hm[i * HS + j] * yv[j];
      p2 = wred_sum(p2);
      if (lane == 0) yv[i] = (rhsv[i] - p2) / Hm[i * HS + i];
      wsync();
    }
    // back substitution: L^T du = y
    for (int i = NU_C - 1; i >= 0; --i) {
      float p2 = 0.f;
      for (int j = i + 1 + lane; j < NU_C; j += 32) p2 += Hm[j * HS + i] * duv[j];
      p2 = wred_sum(p2);
      if (lane == 0) duv[i] = (yv[i] - p2) / Hm[i * HS + i];
      wsync();
    }
    // Gdu, ds, dlam
    for (int j = lane; j < NINEQ_C; j += 32) {
      const float* gr = &sG[j * GS];
      float a = 0.f;
      for (int i = 0; i < NU_C; ++i) a += gr[i] * duv[i];
      const float rpj = rpv[j], sj = sv[j], lj = lv[j];
      const float rc = SIGMA_C * mu - sj * lj;
      dsv[j] = -rpj - a;
      dlv[j] = (rc + lj * (rpj + a)) / sj;
    }
    // alpha = min(1, 0.99*min(max_step(s,ds), max_step(lam,dlam)))
    float am = BIG_C;
    for (int j = lane; j < NINEQ_C; j += 32) {
      const float d_s = dsv[j], d_l = dlv[j];
      const float a1 = (d_s < 0.f) ? (-sv[j] / d_s) : BIG_C;
      const float a2 = (d_l < 0.f) ? (-lv[j] / d_l) : BIG_C;
      am = fminf(am, fminf(a1, a2));
    }
    am = wred_min(am);
    const float alpha = fminf(1.0f, 0.99f * am);
    for (int i = lane; i < NU_C; i += 32) uv[i] += alpha * duv[i];
    for (int j = lane; j < NINEQ_C; j += 32) {
      sv[j] += alpha * dsv[j];
      lv[j] += alpha * dlv[j];
    }
    wsync();
  }

  // Q_value = sum(0.5*(u@Q)*u + p*u) ; u0 = u[0]
  float qv = 0.f;
  for (int i = lane; i < NU_C; i += 32) {
    float uq = 0.f;
    for (int k = 0; k < NU_C; ++k) uq += uv[k] * sQ[k * QS + i];
    qv += (0.5f * uq + pv[i]) * uv[i];
  }
  qv = wred_sum(qv);
  if (lane == 0) {
    out[b] = qv;
    out[BATCH_C + b] = uv[0];
  }
}

// ---------------------------------------------------------------------------
extern "C" void kernel_launch(void* const* d_in, const int* in_sizes, int n_in,
                              void* d_out, int out_size, void* d_ws,
                              size_t ws_size, hipStream_t stream) {
  (void)in_sizes; (void)n_in; (void)out_size; (void)ws_size;
  const float* x  = (const float*)d_in[0];
  const float* W1 = (const float*)d_in[1];
  const float* b1 = (const float*)d_in[2];
  const float* W2 = (const float*)d_in[3];
  const float* b2 = (const float*)d_in[4];
  const float* L  = (const float*)d_in[5];
  const float* LP = (const float*)d_in[6];
  const float* LR = (const float*)d_in[7];
  const float* A  = (const float*)d_in[8];
  const float* Bm = (const float*)d_in[9];
  const float* u0 = (const float*)d_in[10];
  const float* s0 = (const float*)d_in[11];

  float* ws = (float*)d_ws;
  float* qp = ws;                                   // 1600 + 5920 + 148 = 7668
  float* p  = ws + 7668;                            // 4096*40

  build_qp_kernel<<<1, 128, 0, stream>>>(L, LP, LR, A, Bm, u0, s0, qp);

  const size_t mlp_sh = (size_t)(32 * 513 + 32 * NIN_C) * sizeof(float);
  mlp_kernel<<<BATCH_C / 32, 256, mlp_sh, stream>>>(x, W1, b1, W2, b2, p);

  const size_t ipm_sh =
      (size_t)(48 * QS + NINEQ_C * GS + NINEQ_C +
               4 * (48 * HS + 7 * NINEQ_C + 6 * NU_C)) * sizeof(float);
  ipm_kernel<<<BATCH_C / 4, 128, ipm_sh, stream>>>(qp, p, (float*)d_out);
}